// RecurrentCRF_38766374814315
// MI455X (gfx1250) — compile-verified
//
#include <hip/hip_runtime.h>
#include <hip/hip_bf16.h>
#include <stdint.h>

// Problem constants (fixed by setup_inputs)
static constexpr int C  = 4;     // NUM_CLASS
static constexpr int H  = 64;
static constexpr int W  = 512;
static constexpr int HT = 16;    // tile height
static constexpr int WT = 64;    // tile width
static constexpr int HP = HT + 2;   // 18 (halo 1 in H)
static constexpr int WP = WT + 4;   // 68 (halo 2 in W)
static constexpr int PLANE = HP * WP;   // 1224
static constexpr float ANG_COEF = 0.02f;
static constexpr float BI_COEF  = 0.1f;

#ifndef __has_builtin
#define __has_builtin(x) 0
#endif

#if __has_builtin(__builtin_amdgcn_global_load_async_to_lds_b32)
#define USE_ASYNC_LDS 1
#else
#define USE_ASYNC_LDS 0
#endif

#if USE_ASYNC_LDS
typedef __attribute__((address_space(1))) int as1_int;
typedef __attribute__((address_space(3))) int as3_int;
#endif

__device__ __forceinline__ void async_copy_f32(float* lds_dst, const float* g_src) {
#if USE_ASYNC_LDS
  __builtin_amdgcn_global_load_async_to_lds_b32(
      (as1_int*)g_src,     // global source (AS1)
      (as3_int*)lds_dst,   // LDS destination (AS3)
      /*imm offset*/0, /*cpol*/0);
#else
  *lds_dst = *g_src;   // synchronous fallback
#endif
}

__device__ __forceinline__ void wait_async_zero() {
#if USE_ASYNC_LDS
#if __has_builtin(__builtin_amdgcn_s_wait_asynccnt)
  __builtin_amdgcn_s_wait_asynccnt(0);
#else
  asm volatile("s_wait_asynccnt 0" ::: "memory");
#endif
#endif
}

// ---------------------------------------------------------------------------
// Kernel 1: per-row softmax stats.  One block per (b,c,h) row of 512 floats.
// stats[2*row] = rowmax, stats[2*row+1] = 1/sum(exp(x-rowmax))
// ---------------------------------------------------------------------------
__global__ __launch_bounds__(256) void row_stats_kernel(
    const float* __restrict__ cur, float* __restrict__ stats) {
  const int row = blockIdx.x;
  const float* r = cur + (size_t)row * W;
  const int t = threadIdx.x;           // 0..255, each handles 2 elems
  __shared__ float red[256];

  float a = r[t];
  float b = r[t + 256];
  red[t] = fmaxf(a, b);
  __syncthreads();
  for (int s = 128; s > 0; s >>= 1) {
    if (t < s) red[t] = fmaxf(red[t], red[t + s]);
    __syncthreads();
  }
  const float rowmax = red[0];
  __syncthreads();

  red[t] = expf(a - rowmax) + expf(b - rowmax);
  __syncthreads();
  for (int s = 128; s > 0; s >>= 1) {
    if (t < s) red[t] += red[t + s];
    __syncthreads();
  }
  if (t == 0) {
    stats[2 * row]     = rowmax;
    stats[2 * row + 1] = 1.0f / red[0];
  }
}

// ---------------------------------------------------------------------------
// Kernel 2: one CRF iteration step.
//   Block: 256 threads (8 wave32), tile HTxWT of one batch image, 4 classes.
//   Each thread owns 4 consecutive w-pixels -> all filter traffic is B128.
//   LDS: u[C][HP][WP] (raw x -> transformed to unary), msk[HP][WP]. ~24.5 KB.
// ---------------------------------------------------------------------------
__global__ __launch_bounds__(256) void crf_step_kernel(
    const float* __restrict__ cur,    // [B,C,H,W]
    const float* __restrict__ mask,   // [B,1,H,W]
    const float* __restrict__ filt,   // [B,C,14,H,W]
    const float* __restrict__ stats,  // [B*C*H][2]
    float* __restrict__ out) {        // [B,C,H,W]
  __shared__ float u[C][HP][WP];
  __shared__ float msk[HP][WP];
  __shared__ float s_rm[C][HP];
  __shared__ float s_ri[C][HP];

  const int b  = blockIdx.z;
  const int h0 = blockIdx.y * HT;
  const int w0 = blockIdx.x * WT;
  const int tid = threadIdx.x;

  // --- stage per-row softmax stats for this tile's H-range ------------------
  if (tid < C * HP) {
    const int c  = tid / HP;
    const int hh = tid - c * HP;
    const int gh = h0 + hh - 1;
    float rm = 0.0f, ri = 0.0f;
    if (gh >= 0 && gh < H) {
      const int row = (b * C + c) * H + gh;
      rm = stats[2 * row];
      ri = stats[2 * row + 1];
    }
    s_rm[c][hh] = rm;
    s_ri[c][hh] = ri;
  }

  // --- async-stage x tile (+halo) for 4 classes and the mask plane ----------
  // plane 0..3 = classes (raw x), plane 4 = lidar_mask.  OOB -> 0 (zero pad).
  // Each LDS element has exactly one writer (ds-store zero OR async load);
  // visibility: s_wait_asynccnt 0 (own wave) + workgroup barrier.
  const int TOT = PLANE * (C + 1);   // 6120 elements
  for (int i = tid; i < TOT; i += 256) {
    const int pl  = i / PLANE;
    const int rem = i - pl * PLANE;
    const int hh  = rem / WP;
    const int ww  = rem - hh * WP;
    const int gh  = h0 + hh - 1;
    const int gw  = w0 + ww - 2;
    float* laddr = (pl < C) ? &u[pl][hh][ww] : &msk[hh][ww];
    if (gh >= 0 && gh < H && gw >= 0 && gw < W) {
      const float* gaddr = (pl < C)
          ? cur  + (((size_t)(b * C + pl) * H + gh) * W + gw)
          : mask + (((size_t)b * H + gh) * W + gw);
      async_copy_f32(laddr, gaddr);
    } else {
      *laddr = 0.0f;
    }
  }
  wait_async_zero();
  __syncthreads();

  // --- transform staged raw x -> unary (softmax over W) in LDS --------------
  for (int i = tid; i < PLANE * C; i += 256) {
    const int c   = i / PLANE;
    const int rem = i - c * PLANE;
    const int hh  = rem / WP;
    const int ww  = rem - hh * WP;
    const int gh  = h0 + hh - 1;
    const int gw  = w0 + ww - 2;
    if (gh >= 0 && gh < H && gw >= 0 && gw < W) {
      u[c][hh][ww] = expf(u[c][hh][ww] - s_rm[c][hh]) * s_ri[c][hh];
    }
  }
  __syncthreads();

  // --- Gaussian weights (ang kernel == bi_ang kernel: theta identical) ------
  // neighbor order matches reference: dz-major, da-minor, center skipped
  const int NDZ[14] = {0,0,0,0,0, 1,1,1,1, 2,2,2,2,2};
  const int NDA[14] = {0,1,2,3,4, 0,1,3,4, 0,1,2,3,4};
  float kwn[14];
#pragma unroll
  for (int n = 0; n < 14; ++n) {
    const int dz = NDZ[n] - 1;
    const int da = NDA[n] - 2;
    const float d2 = (float)(dz * dz + da * da);
    kwn[n] = expf(d2 * (-1.0f / 1.62f));   // exp(-d2 / (2*0.9^2))
  }

  const size_t HW = (size_t)H * W;

  // --- compute: one quad of 4 consecutive w-pixels per thread ---------------
  const int hh = tid >> 4;            // 0..15
  const int ww = (tid & 15) * 4;      // 0,4,...,60
  const int gh = h0 + hh;
  const int gw = w0 + ww;             // multiple of 4 -> 16B-aligned filters

  const float* fbase = filt + ((((size_t)(b * C) * 14) * H + gh) * W + gw);

  float acc_a[C][4];
  float acc_cd[C][4];
#pragma unroll
  for (int c = 0; c < C; ++c)
#pragma unroll
    for (int j = 0; j < 4; ++j) { acc_a[c][j] = 0.0f; acc_cd[c][j] = 0.0f; }

#pragma unroll
  for (int n = 0; n < 14; ++n) {
    const int zr = hh + NDZ[n];
    const int wc = ww + NDA[n];
    const float kw = kwn[n];
    float mm[4];
#pragma unroll
    for (int j = 0; j < 4; ++j) mm[j] = msk[zr][wc + j];
#pragma unroll
    for (int c = 0; c < C; ++c) {
      const float4 f =
          *reinterpret_cast<const float4*>(fbase + ((size_t)c * 14 + n) * HW);
      const float fv[4] = {f.x, f.y, f.z, f.w};
#pragma unroll
      for (int j = 0; j < 4; ++j) {
        const float uv = u[c][zr][wc + j];
        acc_a[c][j]  = fmaf(kw, uv, acc_a[c][j]);            // depthwise Gauss
        acc_cd[c][j] = fmaf(fv[j], uv * mm[j], acc_cd[c][j]); // bilateral
      }
    }
  }

  // --- finalize: bi = condensed*mask*bi_ang; compat mix; store B128 ---------
  float mc[4];
#pragma unroll
  for (int j = 0; j < 4; ++j) mc[j] = msk[hh + 1][ww + 2 + j];

  float sa[4] = {0.0f, 0.0f, 0.0f, 0.0f};
  float sb[4] = {0.0f, 0.0f, 0.0f, 0.0f};
#pragma unroll
  for (int c = 0; c < C; ++c) {
#pragma unroll
    for (int j = 0; j < 4; ++j) {
      const float a  = acc_a[c][j];
      const float bb = acc_cd[c][j] * mc[j] * a;
      acc_cd[c][j] = bb;
      sa[j] += a;
      sb[j] += bb;
    }
  }
#pragma unroll
  for (int c = 0; c < C; ++c) {
    float o[4];
#pragma unroll
    for (int j = 0; j < 4; ++j) {
      o[j] = u[c][hh + 1][ww + 2 + j]
           + ANG_COEF * (sa[j] - acc_a[c][j])
           + BI_COEF  * (sb[j] - acc_cd[c][j]);
    }
    float4 ov = make_float4(o[0], o[1], o[2], o[3]);
    *reinterpret_cast<float4*>(
        out + (((size_t)(b * C + c) * H) + gh) * W + gw) = ov;
  }
}

// ---------------------------------------------------------------------------
extern "C" void kernel_launch(void* const* d_in, const int* in_sizes, int n_in,
                              void* d_out, int out_size, void* d_ws, size_t ws_size,
                              hipStream_t stream) {
  (void)n_in; (void)out_size; (void)ws_size;
  const float* x    = (const float*)d_in[0];  // [B,C,H,W]
  const float* mask = (const float*)d_in[1];  // [B,1,H,W]
  const float* filt = (const float*)d_in[2];  // [B,C,14,H,W]
  float* out = (float*)d_out;

  const int B = in_sizes[0] / (C * H * W);    // 32
  const int rows = B * C * H;                 // 8192

  float* tmp   = (float*)d_ws;                       // [B,C,H,W]
  float* stats = tmp + (size_t)B * C * H * W;        // [rows*2]

  const dim3 sgrid(W / WT, H / HT, B);        // (8, 4, 32)

  // iter 0: x -> out
  row_stats_kernel<<<rows, 256, 0, stream>>>(x, stats);
  crf_step_kernel<<<sgrid, 256, 0, stream>>>(x, mask, filt, stats, out);
  // iter 1: out -> tmp
  row_stats_kernel<<<rows, 256, 0, stream>>>(out, stats);
  crf_step_kernel<<<sgrid, 256, 0, stream>>>(out, mask, filt, stats, tmp);
  // iter 2: tmp -> out
  row_stats_kernel<<<rows, 256, 0, stream>>>(tmp, stats);
  crf_step_kernel<<<sgrid, 256, 0, stream>>>(tmp, mask, filt, stats, out);
}